// HeteroGNN_73272142069881
// MI455X (gfx1250) — compile-verified
//
#include <hip/hip_runtime.h>

// HeteroGNN (bipartite SAGE x2 + edge predictor) for MI455X / gfx1250.
// Algebraic refactor: project-then-segment-mean (linearity of mean), and
// decomposed edge predictor (per-node dots + gather-add).
// Dense GEMMs: V_WMMA_F32_16X16X4_F32 (full f32 precision, wave32), with the
// B operand staged in LDS (bank-conflict-free padded layout) per workgroup.

#define NL 100000   // ligand nodes
#define NT 4000     // target nodes
#define DL 4        // ligand feature dim
#define DT 1280     // target feature dim
#define H  128      // hidden dim

#define KC   128    // K-chunk staged in LDS
#define PADN 136    // padded LDS row stride (floats): halves hit disjoint banks

typedef float v2f __attribute__((ext_vector_type(2)));
typedef float v8f __attribute__((ext_vector_type(8)));

// ---------------------------------------------------------------------------
// WMMA f32 16x16x4 lane layout (ISA 7.12.2, 32-bit A 16x4 / B 4x16):
//  A: lanes 0-15: M=lane, v[0]=K0,v[1]=K1 ; lanes 16-31: M=lane-16, K2/K3.
//  B: mirrored with N=lane&15, halves select K rows {0,1} / {2,3}.
//  C/D: v[i] = row (i + 8*(lane>=16)), col lane&15.
// ---------------------------------------------------------------------------

// Accumulate acc += A[mBase:+16, 0:K] @ B[0:K, 0:128] for this wave's N-tile,
// staging B in LDS chunk by chunk (cooperative across the whole block).
__device__ inline void gemm_chunk_lds(const float* __restrict__ A,
                                      const float* __restrict__ B,
                                      int K, float* __restrict__ ldsB,
                                      int mBase, int r, int half, int nt,
                                      v8f& acc) {
  const int tid = threadIdx.x;
  const float* arow = A + (size_t)(mBase + r) * K;
  const int col = nt * 16 + r;
  for (int k0 = 0; k0 < K; k0 += KC) {
    __syncthreads();                       // protect LDS from previous chunk
    for (int idx = tid; idx < KC * H; idx += blockDim.x) {
      const int kr = idx >> 7, c = idx & (H - 1);
      ldsB[kr * PADN + c] = B[(size_t)(k0 + kr) * H + c];
    }
    __syncthreads();
#pragma unroll 4
    for (int k = 0; k < KC; k += 4) {
      const int ka = k + 2 * half;
      v2f a = *(const v2f*)(arow + k0 + ka);   // 8B-aligned (ka even)
      v2f b;
      b[0] = ldsB[ka * PADN + col];            // ds_load, conflict-free
      b[1] = ldsB[(ka + 1) * PADN + col];
      acc = __builtin_amdgcn_wmma_f32_16x16x4_f32(false, a, false, b,
                                                  (short)0, acc, false, false);
    }
  }
}

// C[M,128] = act( A@B (+ A2@B2) (+ bias) ).  One block per 16-row M-tile;
// 8 waves = the 8 N-tiles of N=128. All waves always active -> EXEC all-ones.
__global__ void __launch_bounds__(256)
wmma_gemm_lds(const float* __restrict__ A, const float* __restrict__ B,
              const float* __restrict__ A2, const float* __restrict__ B2,
              const float* __restrict__ bias, float* __restrict__ C,
              int K, int K2, int doRelu) {
  __shared__ float ldsB[KC * PADN];          // 68 KB of the 320 KB WGP LDS
  const int mBase = blockIdx.x * 16;
  const int nt = threadIdx.x >> 5;           // wave id = N-tile
  const int l = threadIdx.x & 31;
  const int r = l & 15;
  const int half = l >> 4;

  v8f acc = {};
  gemm_chunk_lds(A, B, K, ldsB, mBase, r, half, nt, acc);
  if (A2) gemm_chunk_lds(A2, B2, K2, ldsB, mBase, r, half, nt, acc);

  const int col = nt * 16 + r;
  const float bv = bias ? bias[col] : 0.0f;
#pragma unroll
  for (int i = 0; i < 8; ++i) {
    const int row = mBase + i + half * 8;
    float v = acc[i] + bv;
    if (doRelu) v = fmaxf(v, 0.0f);
    C[(size_t)row * H + col] = v;
  }
}

// ---------------------------------------------------------------------------
__global__ void zero_f32(float* __restrict__ p, int n) {
  for (int i = blockIdx.x * blockDim.x + threadIdx.x; i < n;
       i += gridDim.x * blockDim.x)
    p[i] = 0.0f;
}

// conv1 edge pass: degree counts + 4-dim ligand feature scatter (target side).
__global__ void scatter1(const float* __restrict__ xl,
                         const int* __restrict__ src, const int* __restrict__ dst,
                         float* __restrict__ mt4,
                         unsigned* __restrict__ cntT, unsigned* __restrict__ cntL,
                         int E) {
  for (int e = blockIdx.x * blockDim.x + threadIdx.x; e < E;
       e += gridDim.x * blockDim.x) {
    const int s = src[e], d = dst[e];
    atomicAdd(&cntT[d], 1u);
    atomicAdd(&cntL[s], 1u);
#pragma unroll
    for (int j = 0; j < DL; ++j)
      atomicAdd(&mt4[d * DL + j], xl[(size_t)s * DL + j]);
  }
}

// conv1 ligand side: accL[src] += (x_target @ W1tl_l)[dst]  (project-then-mean)
__global__ void scatter_proj(const float* __restrict__ pt1,
                             const int* __restrict__ src,
                             const int* __restrict__ dst,
                             float* __restrict__ accL, int EH) {
  for (int i = blockIdx.x * blockDim.x + threadIdx.x; i < EH;
       i += gridDim.x * blockDim.x) {
    const int e = i >> 7, k = i & (H - 1);
    atomicAdd(&accL[(size_t)src[e] * H + k], pt1[(size_t)dst[e] * H + k]);
  }
}

// conv2 edge pass: accT2[dst] += h_l[src];  accL2[src] += h_t[dst]
__global__ void scatter2(const float* __restrict__ hl,
                         const float* __restrict__ ht,
                         const int* __restrict__ src,
                         const int* __restrict__ dst,
                         float* __restrict__ accT2, float* __restrict__ accL2,
                         int EH) {
  for (int i = blockIdx.x * blockDim.x + threadIdx.x; i < EH;
       i += gridDim.x * blockDim.x) {
    const int e = i >> 7, k = i & (H - 1);
    const int s = src[e], d = dst[e];
    atomicAdd(&accT2[(size_t)d * H + k], hl[(size_t)s * H + k]);
    atomicAdd(&accL2[(size_t)s * H + k], ht[(size_t)d * H + k]);
  }
}

// h_l = relu(accL/cnt + b1tl + x_ligand @ W1tl_r)   (in place on accL)
__global__ void fin_hl(float* __restrict__ accL, const unsigned* __restrict__ cntL,
                       const float* __restrict__ xl,
                       const float* __restrict__ W1tl_r,
                       const float* __restrict__ b1tl, int n) {
  for (int i = blockIdx.x * blockDim.x + threadIdx.x; i < n;
       i += gridDim.x * blockDim.x) {
    const int row = i >> 7, k = i & (H - 1);
    const float c = fmaxf((float)cntL[row], 1.0f);
    float v = accL[i] / c + b1tl[k];
#pragma unroll
    for (int j = 0; j < DL; ++j)
      v += xl[(size_t)row * DL + j] * W1tl_r[j * H + k];
    accL[i] = fmaxf(v, 0.0f);
  }
}

// h_t = relu((mt4/cnt) @ W1lt_l + b1lt + ht_pre)   (in place on ht)
__global__ void fin_ht(float* __restrict__ ht, const float* __restrict__ mt4,
                       const unsigned* __restrict__ cntT,
                       const float* __restrict__ W1lt_l,
                       const float* __restrict__ b1lt, int n) {
  for (int i = blockIdx.x * blockDim.x + threadIdx.x; i < n;
       i += gridDim.x * blockDim.x) {
    const int row = i >> 7, k = i & (H - 1);
    const float c = fmaxf((float)cntT[row], 1.0f);
    float v = ht[i] + b1lt[k];
#pragma unroll
    for (int j = 0; j < DL; ++j)
      v += (mt4[row * DL + j] / c) * W1lt_l[j * H + k];
    ht[i] = fmaxf(v, 0.0f);
  }
}

// mean finalize for conv2 accumulators: acc[row,:] /= max(cnt[row],1)
__global__ void div_rows(float* __restrict__ acc, const unsigned* __restrict__ cnt,
                         int n) {
  for (int i = blockIdx.x * blockDim.x + threadIdx.x; i < n;
       i += gridDim.x * blockDim.x)
    acc[i] /= fmaxf((float)cnt[i >> 7], 1.0f);
}

// out[row] = dot(Mx[row, 0:H], w[0:H])   — one wave32 per row
__global__ void rowdot(const float* __restrict__ Mx, const float* __restrict__ w,
                       float* __restrict__ out, int rows) {
  const int wave = (blockIdx.x * blockDim.x + threadIdx.x) >> 5;
  const int l = threadIdx.x & 31;
  if (wave >= rows) return;
  const float* rp = Mx + (size_t)wave * H;
  float s = 0.0f;
#pragma unroll
  for (int k = l; k < H; k += 32) s += rp[k] * w[k];
#pragma unroll
  for (int off = 16; off > 0; off >>= 1) s += __shfl_xor(s, off, 32);
  if (l == 0) out[wave] = s;
}

// out[e] = p_l[src[e]] + p_t[dst[e]] + bp
__global__ void edge_out(const float* __restrict__ pl, const float* __restrict__ pt,
                         const int* __restrict__ src, const int* __restrict__ dst,
                         const float* __restrict__ bp, float* __restrict__ out,
                         int E) {
  for (int e = blockIdx.x * blockDim.x + threadIdx.x; e < E;
       e += gridDim.x * blockDim.x)
    out[e] = pl[src[e]] + pt[dst[e]] + bp[0];
}

// ---------------------------------------------------------------------------
static inline int cdiv(long long a, int b) { return (int)((a + b - 1) / b); }

extern "C" void kernel_launch(void* const* d_in, const int* in_sizes, int n_in,
                              void* d_out, int out_size, void* d_ws, size_t ws_size,
                              hipStream_t stream) {
  const float* xl      = (const float*)d_in[0];
  const float* xt      = (const float*)d_in[1];
  const int*   ei      = (const int*)  d_in[2];
  const int    E       = in_sizes[2] / 2;
  const int*   src     = ei;           // ligand indices
  const int*   dst     = ei + E;       // target indices
  const float* W1lt_l  = (const float*)d_in[3];
  const float* b1lt    = (const float*)d_in[4];
  const float* W1lt_r  = (const float*)d_in[5];
  const float* W1tl_l  = (const float*)d_in[6];
  const float* b1tl    = (const float*)d_in[7];
  const float* W1tl_r  = (const float*)d_in[8];
  const float* W2lt_l  = (const float*)d_in[9];
  const float* b2lt    = (const float*)d_in[10];
  const float* W2lt_r  = (const float*)d_in[11];
  const float* W2tl_l  = (const float*)d_in[12];
  const float* b2tl    = (const float*)d_in[13];
  const float* W2tl_r  = (const float*)d_in[14];
  const float* Wp      = (const float*)d_in[15];
  const float* bp      = (const float*)d_in[16];
  float*       out     = (float*)d_out;

  // Workspace layout (~163 MB of f32)
  float* ws = (float*)d_ws;
  size_t off = 0;
  float*    accL  = ws + off; off += (size_t)NL * H;  // conv1 lig acc -> h_l
  float*    accL2 = ws + off; off += (size_t)NL * H;  // conv2 lig acc -> m_l2
  float*    hl2   = ws + off; off += (size_t)NL * H;
  float*    accT2 = ws + off; off += (size_t)NT * H;  // conv2 tgt acc -> m_t2
  float*    ht    = ws + off; off += (size_t)NT * H;  // xt@W1lt_r -> h_t
  float*    ht2   = ws + off; off += (size_t)NT * H;
  float*    pt1   = ws + off; off += (size_t)NT * H;  // xt@W1tl_l
  float*    mt4   = ws + off; off += (size_t)NT * DL;
  unsigned* cntT  = (unsigned*)(ws + off); off += NT;
  unsigned* cntL  = (unsigned*)(ws + off); off += NL;
  float*    pl    = ws + off; off += NL;
  float*    ptv   = ws + off; off += NT;

  const int TB = 256;

  // 0) zero accumulators (accL+accL2 contiguous; accT2; mt4+cntT+cntL contiguous)
  {
    int n0 = 2 * NL * H;
    zero_f32<<<cdiv(n0, TB), TB, 0, stream>>>(accL, n0);
    int n1 = NT * H;
    zero_f32<<<cdiv(n1, TB), TB, 0, stream>>>(accT2, n1);
    int n2 = NT * DL + NT + NL;
    zero_f32<<<cdiv(n2, TB), TB, 0, stream>>>(mt4, n2);
  }

  // 1) conv1 edge pass: degrees + 4-dim target-side scatter
  scatter1<<<cdiv(E, TB), TB, 0, stream>>>(xl, src, dst, mt4, cntT, cntL, E);

  // 2) target projections (M=4000, K=1280, N=128) — WMMA f32, B staged in LDS
  wmma_gemm_lds<<<NT / 16, TB, 0, stream>>>(xt, W1tl_l, nullptr, nullptr,
                                            nullptr, pt1, DT, 0, 0);
  wmma_gemm_lds<<<NT / 16, TB, 0, stream>>>(xt, W1lt_r, nullptr, nullptr,
                                            nullptr, ht, DT, 0, 0);

  // 3) ligand-side projected scatter: accL[src] += pt1[dst]
  scatter_proj<<<cdiv((long long)E * H, TB), TB, 0, stream>>>(pt1, src, dst,
                                                              accL, E * H);

  // 4) finalize conv1
  fin_ht<<<cdiv(NT * H, TB), TB, 0, stream>>>(ht, mt4, cntT, W1lt_l, b1lt, NT * H);
  fin_hl<<<cdiv(NL * H, TB), TB, 0, stream>>>(accL, cntL, xl, W1tl_r, b1tl, NL * H);

  // 5) conv2 edge pass (both directions) + mean finalize
  scatter2<<<cdiv((long long)E * H, TB), TB, 0, stream>>>(accL, ht, src, dst,
                                                          accT2, accL2, E * H);
  div_rows<<<cdiv(NT * H, TB), TB, 0, stream>>>(accT2, cntT, NT * H);
  div_rows<<<cdiv(NL * H, TB), TB, 0, stream>>>(accL2, cntL, NL * H);

  // 6) conv2 fused dual GEMMs: relu(m@Wl + h@Wr + b) — WMMA f32, LDS-staged B
  wmma_gemm_lds<<<NT / 16, TB, 0, stream>>>(accT2, W2lt_l, ht, W2lt_r,
                                            b2lt, ht2, H, H, 1);
  wmma_gemm_lds<<<NL / 16, TB, 0, stream>>>(accL2, W2tl_l, accL, W2tl_r,
                                            b2tl, hl2, H, H, 1);

  // 7) decomposed edge predictor
  rowdot<<<cdiv((long long)NL * 32, TB), TB, 0, stream>>>(hl2, Wp, pl, NL);
  rowdot<<<cdiv((long long)NT * 32, TB), TB, 0, stream>>>(ht2, Wp + H, ptv, NT);
  edge_out<<<cdiv(E, TB), TB, 0, stream>>>(pl, ptv, src, dst, bp, out, E);
}